// LTI2d_27101243638415
// MI455X (gfx1250) — compile-verified
//
#include <hip/hip_runtime.h>

// ---------------------------------------------------------------------------
// LTI2d on MI455X (gfx1250).
// FFT-bandwidth-bound pipeline (~6-8 GB HBM traffic @ 23.3 TB/s); the only
// dense matrix op is the per-frequency 16x16 complex channel mix, done in
// fp32 with V_WMMA_F32_16X16X4_F32 (fp32 required: FFT data, bf16 would
// destroy accuracy).  Real expansion per frequency:
//   [[Xr,-Xi],[Xi,Xr]](8x32) x [Hr;Hi](32x16)  ->  M=8,K=32,N=16
//   = 8 chained K=4 fp32 WMMAs per wave (full N, 50% M).
// H tile per frequency is DMA'd into LDS by the Tensor Data Mover.
// Workspace: R0 (270MB) | R1 (538MB) | R2 (1076MB)  ~= 1.88 GB total.
// ---------------------------------------------------------------------------

#define DEV __device__ __forceinline__
typedef long long i64;

constexpr int   Bb  = 4;
constexpr int   L   = 512;
constexpr int   N2  = 1024;
constexpr int   NF  = 513;   // rfft bins at 1024
constexpr int   NF5 = 257;   // rfft bins at 512
constexpr int   D   = 16;
constexpr float PIf = 3.14159265358979323846f;

struct c2 { float x, y; };
DEV c2 cmul(c2 a, c2 b) { return { a.x * b.x - a.y * b.y, a.x * b.y + a.y * b.x }; }

// Radix-2 Stockham autosort FFT in LDS, double-buffered. blockDim.x threads.
template <int N>
DEV c2* fft_stockham(c2* bufA, c2* bufB, float sign) {
    c2* s = bufA;
    c2* d = bufB;
    int l = N >> 1, m = 1;
    while (l >= 1) {
        for (int t = threadIdx.x; t < (N >> 1); t += blockDim.x) {
            int j = t / m;
            int k = t - j * m;
            float ang = sign * (PIf * (float)j / (float)l);
            float sn, cs;
            __sincosf(ang, &sn, &cs);
            c2 w  = { cs, sn };
            c2 c0 = s[k + j * m];
            c2 c1 = s[k + j * m + l * m];
            c2 su = { c0.x + c1.x, c0.y + c1.y };
            c2 df = { c0.x - c1.x, c0.y - c1.y };
            d[k + 2 * j * m]     = su;
            d[k + 2 * j * m + m] = cmul(w, df);
        }
        __syncthreads();
        c2* tmp = s; s = d; d = tmp;
        l >>= 1; m <<= 1;
    }
    return s;
}

// --- K1: Hc(u,v) = B(u,v)/A(u,v), a/b are 9x9 padded taps ----------------
__global__ void k_eval_rational(const float* __restrict__ num0,
                                const float* __restrict__ den,
                                c2* __restrict__ Hc) {
    i64 idx = (i64)blockIdx.x * blockDim.x + threadIdx.x;
    const i64 total = (i64)D * D * L * NF5;
    if (idx >= total) return;
    int v = (int)(idx % NF5);
    i64 t = idx / NF5;
    int u  = (int)(t % L);
    int ij = (int)(t / L);
    c2 wu, wv;
    __sincosf(-2.f * PIf * (float)u / (float)L, &wu.y, &wu.x);
    __sincosf(-2.f * PIf * (float)v / (float)L, &wv.y, &wv.x);
    c2 Aa = { 0.f, 0.f }, Bv = { 0.f, 0.f };
    c2 pu = { 1.f, 0.f };
    for (int p = 0; p < 9; ++p) {
        c2 pv = { 1.f, 0.f };
        for (int q = 0; q < 9; ++q) {
            c2 e = cmul(pu, pv);
            float aco = (p == 0 || q == 0) ? 1.f : den[(i64)ij * 64 + (p - 1) * 8 + (q - 1)];
            float bco = (p == 0 || q == 0) ? 0.f : num0[(i64)ij * 64 + (p - 1) * 8 + (q - 1)];
            Aa.x += aco * e.x; Aa.y += aco * e.y;
            Bv.x += bco * e.x; Bv.y += bco * e.y;
            pv = cmul(pv, wv);
        }
        pu = cmul(pu, wu);
    }
    float d2 = Aa.x * Aa.x + Aa.y * Aa.y + 1e-30f;
    Hc[idx] = { (Bv.x * Aa.x + Bv.y * Aa.y) / d2,
                (Bv.y * Aa.x - Bv.x * Aa.y) / d2 };
}

// --- K2a: inverse FFT512 along u for each (ij, v) ------------------------
__global__ void k_h_col_ifft512(const c2* __restrict__ Hc, c2* __restrict__ T1) {
    __shared__ c2 A[512], Bf[512];
    int v  = blockIdx.x % NF5;
    int ij = blockIdx.x / NF5;
    const c2* src = Hc + (i64)ij * L * NF5;
    for (int u = threadIdx.x; u < L; u += blockDim.x) A[u] = src[(i64)u * NF5 + v];
    __syncthreads();
    c2* r = fft_stockham<512>(A, Bf, +1.f);
    c2* dst = T1 + (i64)ij * L * NF5;
    const float sc = 1.f / 512.f;
    for (int u = threadIdx.x; u < L; u += blockDim.x) {
        c2 z = r[u];
        dst[(i64)u * NF5 + v] = { z.x * sc, z.y * sc };
    }
}

// --- K2b: inverse rFFT512 along v (Hermitian extend) -> h_trunc ----------
__global__ void k_h_row_irfft512(const c2* __restrict__ T1, float* __restrict__ ht) {
    __shared__ c2 A[512], Bf[512];
    int u  = blockIdx.x % L;
    int ij = blockIdx.x / L;
    const c2* row = T1 + ((i64)ij * L + u) * NF5;
    for (int v = threadIdx.x; v < 512; v += blockDim.x) {
        c2 z;
        if (v <= 256) z = row[v];
        else { c2 w = row[512 - v]; z = { w.x, -w.y }; }
        A[v] = z;
    }
    __syncthreads();
    c2* r = fft_stockham<512>(A, Bf, +1.f);
    float* dst = ht + ((i64)ij * L + u) * L;
    const float sc = 1.f / 512.f;
    for (int w = threadIdx.x; w < L; w += blockDim.x) dst[w] = r[w].x * sc;
}

// --- K3a: forward FFT1024 rows of h_trunc (zero-padded) ------------------
__global__ void k_h_row_fft1024(const float* __restrict__ ht, c2* __restrict__ T2) {
    __shared__ c2 A[1024], Bf[1024];
    int u  = blockIdx.x % L;
    int ij = blockIdx.x / L;
    const float* row = ht + ((i64)ij * L + u) * L;
    for (int w = threadIdx.x; w < N2; w += blockDim.x)
        A[w] = (w < L) ? c2{ row[w], 0.f } : c2{ 0.f, 0.f };
    __syncthreads();
    c2* r = fft_stockham<1024>(A, Bf, -1.f);
    c2* dst = T2 + ((i64)ij * L + u) * NF;
    for (int v = threadIdx.x; v < NF; v += blockDim.x) dst[v] = r[v];
}

// --- K3b: forward FFT1024 cols -> H planes [h][v][ij], +h0 bias ----------
__global__ void k_h_col_fft1024(const c2* __restrict__ T2, const float* __restrict__ h0,
                                float* __restrict__ Hre, float* __restrict__ Him) {
    __shared__ c2 A[1024], Bf[1024];
    int v  = blockIdx.x % NF;
    int ij = blockIdx.x / NF;
    for (int h = threadIdx.x; h < N2; h += blockDim.x)
        A[h] = (h < L) ? T2[((i64)ij * L + h) * NF + v] : c2{ 0.f, 0.f };
    __syncthreads();
    c2* r = fft_stockham<1024>(A, Bf, -1.f);
    float bias = h0[ij];
    for (int h = threadIdx.x; h < N2; h += blockDim.x) {
        i64 o = ((i64)h * NF + v) * 256 + ij;
        Hre[o] = r[h].x + bias;
        Him[o] = r[h].y;
    }
}

// --- K4a: forward FFT1024 rows of x --------------------------------------
__global__ void k_x_row_fft(const float* __restrict__ x, c2* __restrict__ Xrow) {
    __shared__ c2 A[1024], Bf[1024];
    int i = blockIdx.x % D;
    int t = blockIdx.x / D;
    int h = t % L;
    int b = t / L;
    const float* row = x + (((i64)b * L + h) * L) * D + i;
    for (int w = threadIdx.x; w < N2; w += blockDim.x)
        A[w] = (w < L) ? c2{ row[(i64)w * D], 0.f } : c2{ 0.f, 0.f };
    __syncthreads();
    c2* r = fft_stockham<1024>(A, Bf, -1.f);
    for (int v = threadIdx.x; v < NF; v += blockDim.x)
        Xrow[(((i64)b * L + h) * NF + v) * D + i] = r[v];
}

// --- K4b: forward FFT1024 cols of x -> X planes [h][v][b*16+i] -----------
__global__ void k_x_col_fft(const c2* __restrict__ Xrow,
                            float* __restrict__ Xsre, float* __restrict__ Xsim) {
    __shared__ c2 A[1024], Bf[1024];
    int i = blockIdx.x % D;
    int t = blockIdx.x / D;
    int v = t % NF;
    int b = t / NF;
    for (int h = threadIdx.x; h < N2; h += blockDim.x)
        A[h] = (h < L) ? Xrow[(((i64)b * L + h) * NF + v) * D + i] : c2{ 0.f, 0.f };
    __syncthreads();
    c2* r = fft_stockham<1024>(A, Bf, -1.f);
    for (int h = threadIdx.x; h < N2; h += blockDim.x) {
        i64 o = ((i64)h * NF + v) * 64 + b * D + i;
        Xsre[o] = r[h].x;
        Xsim[o] = r[h].y;
    }
}

// --- K5: per-frequency complex channel mix via fp32 WMMA -----------------
typedef __attribute__((ext_vector_type(2))) float v2f;
typedef __attribute__((ext_vector_type(8))) float v8f;
typedef __attribute__((ext_vector_type(4))) unsigned int u32x4;
typedef __attribute__((ext_vector_type(4))) int  i32x4;
typedef __attribute__((ext_vector_type(8))) int  i32x8;

#if defined(__has_builtin)
#if __has_builtin(__builtin_amdgcn_tensor_load_to_lds) && __has_builtin(__builtin_amdgcn_s_wait_tensorcnt)
#define HAVE_TDM 1
#else
#define HAVE_TDM 0
#endif
#else
#define HAVE_TDM 0
#endif

DEV unsigned rfl(unsigned v) { return (unsigned)__builtin_amdgcn_readfirstlane((int)v); }

#if HAVE_TDM
// DMA one 256-float row tile (1 KB) from global into LDS via the TDM.
DEV void tdm_row_to_lds(const float* gsrc, unsigned lds_off) {
    unsigned long long ga = (unsigned long long)gsrc;
    // D# group 0: count=1, lds_addr, 57-bit global addr, type=2 ("image")
    u32x4 g0;
    g0.x = rfl(1u);                                   // count = 1 valid descriptor
    g0.y = rfl(lds_off);                              // lds_addr (bytes)
    g0.z = rfl((unsigned)(ga & 0xffffffffu));         // global_addr[31:0]
    g0.w = rfl((unsigned)(((ga >> 32) & 0x01ffffffu)) | (2u << 30)); // [56:32] | type=2
    // D# group 1: data_size=4B; tensor_dim0=256, tensor_dim1=1;
    //             tile_dim0=256, tile_dim1=1; dim0_stride=256
    i32x8 g1;
    g1[0] = (int)rfl(2u << 16);        // data_size = 2 (4 bytes)
    g1[1] = (int)rfl(256u << 16);      // tensor_dim0[15:0] @ bit48
    g1[2] = (int)rfl(1u << 16);        // tensor_dim0 hi = 0 | tensor_dim1 lo = 1
    g1[3] = (int)rfl(256u << 16);      // tensor_dim1 hi = 0 | tile_dim0 = 256
    g1[4] = (int)rfl(1u);              // tile_dim1 = 1, tile_dim2 = 0
    g1[5] = (int)rfl(256u);            // tensor_dim0_stride lo
    g1[6] = 0;
    g1[7] = 0;
    i32x4 z4 = { 0, 0, 0, 0 };
#if defined(__clang_major__) && (__clang_major__ >= 23)
    i32x8 z8 = { 0, 0, 0, 0, 0, 0, 0, 0 };
    __builtin_amdgcn_tensor_load_to_lds(g0, g1, z4, z4, z8, 0);
#else
    __builtin_amdgcn_tensor_load_to_lds(g0, g1, z4, z4, 0);
#endif
}
#endif

__global__ void k_mix_wmma(const float* __restrict__ Xsre, const float* __restrict__ Xsim,
                           const float* __restrict__ Hre,  const float* __restrict__ Him,
                           float* __restrict__ Ysre, float* __restrict__ Ysim) {
    __shared__ float sX[8][128];  // per wave: [0..63]=Xr[b*16+i], [64..127]=Xi
    __shared__ float sH[8][512];  // per wave: [kk*16+j], kk<16 -> Hr, kk>=16 -> Hi
    int lane = threadIdx.x & 31;
    int wv   = threadIdx.x >> 5;
    i64 f = (i64)blockIdx.x * 8 + wv;   // 1024*513 total, exact multiple of 8

    // prefetch next block's H tile into L2 (global_prefetch_b8)
    __builtin_prefetch(Hre + (f + 8) * 256 + lane, 0, 1);

    // stage H: Tensor Data Mover DMA (2 x 1KB row tiles), fallback plain loads
#if HAVE_TDM
    {
        unsigned base = (unsigned)((unsigned long long)(&sH[wv][0]) & 0xfffffu);
        tdm_row_to_lds(Hre + f * 256, base);
        tdm_row_to_lds(Him + f * 256, base + 256u * 4u);
    }
#else
    #pragma unroll
    for (int t = 0; t < 8; ++t) {
        sH[wv][t * 32 + lane]       = Hre[f * 256 + t * 32 + lane];
        sH[wv][256 + t * 32 + lane] = Him[f * 256 + t * 32 + lane];
    }
#endif

    // stage X with plain coalesced loads (overlaps the TDM transfers)
    sX[wv][lane]           = Xsre[f * 64 + lane];
    sX[wv][lane + 32]      = Xsre[f * 64 + lane + 32];
    sX[wv][64 + lane]      = Xsim[f * 64 + lane];
    sX[wv][64 + lane + 32] = Xsim[f * 64 + lane + 32];

#if HAVE_TDM
    __builtin_amdgcn_s_wait_tensorcnt(0);
#endif
    __syncthreads();

    int m    = lane & 15;             // A row (lanes 16-31 mirror rows for K=2,3)
    int n    = lane & 15;             // B column
    int ksel = (lane >> 4) << 1;      // VGPR half selects K offset {0,2}
    int b_   = m & 3;
    int up   = (m >> 2) & 1;
    float validf = (m < 8) ? 1.f : 0.f;   // rows 8..15 of A are zero padding

    // Branchless fragment gather: A(8x32)=[[Xr,-Xi],[Xi,Xr]]  B(32x16)=[Hr;Hi]
    float aF[16], bF[16];
    #pragma unroll
    for (int c = 0; c < 8; ++c) {
        #pragma unroll
        for (int e = 0; e < 2; ++e) {
            int kk   = 4 * c + ksel + e;
            int part = kk >> 4;               // 0: first 16 K-cols, 1: second
            int i2   = kk & 15;
            int src  = up ^ part;             // 0 -> Xr block, 1 -> Xi block
            float sg = ((part == 1) && (up == 0)) ? -validf : validf;
            aF[c * 2 + e] = sX[wv][src * 64 + b_ * 16 + i2] * sg;
            bF[c * 2 + e] = sH[wv][kk * 16 + n];
        }
    }

    // 8 back-to-back K=4 fp32 WMMAs (accumulate chain on C)
    v8f acc = { 0.f, 0.f, 0.f, 0.f, 0.f, 0.f, 0.f, 0.f };
    #pragma unroll
    for (int c = 0; c < 8; ++c) {
        v2f a  = { aF[c * 2], aF[c * 2 + 1] };
        v2f bm = { bF[c * 2], bF[c * 2 + 1] };
        acc = __builtin_amdgcn_wmma_f32_16x16x4_f32(
                  false, a, false, bm, (short)0, acc, false, false);
    }

    if (lane < 16) {                  // rows 0..3 = Yr[b][j], rows 4..7 = Yi[b][j]
        #pragma unroll
        for (int r = 0; r < 8; ++r) {
            float vv = acc[r];
            if (r < 4) Ysre[f * 64 + r * 16 + lane]       = vv;
            else       Ysim[f * 64 + (r - 4) * 16 + lane] = vv;
        }
    }
}

// --- K6: inverse FFT1024 columns of Y, keep h < 512 ----------------------
__global__ void k_y_col_ifft(const float* __restrict__ Ysre, const float* __restrict__ Ysim,
                             c2* __restrict__ Ycol) {
    __shared__ c2 A[1024], Bf[1024];
    int j = blockIdx.x % D;
    int t = blockIdx.x / D;
    int v = t % NF;
    int b = t / NF;
    for (int h = threadIdx.x; h < N2; h += blockDim.x) {
        i64 o = ((i64)h * NF + v) * 64 + b * D + j;
        A[h] = { Ysre[o], Ysim[o] };
    }
    __syncthreads();
    c2* r = fft_stockham<1024>(A, Bf, +1.f);
    const float sc = 1.f / 1024.f;
    for (int h = threadIdx.x; h < L; h += blockDim.x) {
        c2 z = r[h];
        Ycol[(((i64)b * L + h) * NF + v) * D + j] = { z.x * sc, z.y * sc };
    }
}

// --- K7: inverse rFFT1024 rows of Y (Hermitian extend), crop w<512 -------
__global__ void k_y_row_ifft(const c2* __restrict__ Ycol, float* __restrict__ out) {
    __shared__ c2 A[1024], Bf[1024];
    int j = blockIdx.x % D;
    int t = blockIdx.x / D;
    int h = t % L;
    int b = t / L;
    const c2* row = Ycol + (((i64)b * L + h) * NF) * D + j;   // stride D
    for (int v = threadIdx.x; v < N2; v += blockDim.x) {
        c2 z;
        if (v <= 512) z = row[(i64)v * D];
        else { c2 w = row[(i64)(N2 - v) * D]; z = { w.x, -w.y }; }
        A[v] = z;
    }
    __syncthreads();
    c2* r = fft_stockham<1024>(A, Bf, +1.f);
    const float sc = 1.f / 1024.f;
    for (int w = threadIdx.x; w < L; w += blockDim.x)
        out[(((i64)b * L + h) * L + w) * D + j] = r[w].x * sc;
}

// ---------------------------------------------------------------------------
extern "C" void kernel_launch(void* const* d_in, const int* in_sizes, int n_in,
                              void* d_out, int out_size, void* d_ws, size_t ws_size,
                              hipStream_t stream) {
    const float* x    = (const float*)d_in[0];  // [4,512,512,16]
    const float* h0   = (const float*)d_in[1];  // [16,16,1,1]
    const float* num  = (const float*)d_in[2];  // [4,16,16,8,8] (only set 0 used)
    const float* den  = (const float*)d_in[3];  // [16,16,8,8]
    float* out = (float*)d_out;

    // workspace regions (sequential kernel chain makes aliasing safe)
    const i64 SZ_R0    = (i64)256 * L * NF5 * sizeof(c2);       // 269,484,032
    const i64 SZ_R1    = (i64)256 * L * NF  * sizeof(c2);       // 537,919,488
    const i64 SZ_PLANE = (i64)N2 * NF * 256 * sizeof(float);    // 537,919,488
    const i64 SZ_XP    = (i64)N2 * NF * 64  * sizeof(float);    // 134,479,872
    char* w  = (char*)d_ws;
    char* R0 = w;
    char* R1 = w + SZ_R0;
    char* R2 = w + SZ_R0 + SZ_R1;                 // needs ~1.88 GB total
    (void)ws_size; (void)n_in; (void)in_sizes; (void)out_size;

    c2*    Hc   = (c2*)R0;
    c2*    T1   = (c2*)R1;
    float* ht   = (float*)R0;
    c2*    T2   = (c2*)R1;
    float* Hre  = (float*)R2;
    float* Him  = (float*)(R2 + SZ_PLANE);
    c2*    Xrow = (c2*)R0;
    float* Xsre = (float*)R1;
    float* Xsim = (float*)(R1 + SZ_XP);
    float* Ysre = (float*)R0;
    float* Ysim = (float*)(R0 + SZ_XP);
    c2*    Ycol = (c2*)R1;

    const int T = 256;

    // H path
    {
        i64 total = (i64)D * D * L * NF5;
        int blocks = (int)((total + T - 1) / T);
        k_eval_rational<<<blocks, T, 0, stream>>>(num, den, Hc);
    }
    k_h_col_ifft512 <<<256 * NF5, T, 0, stream>>>(Hc, T1);
    k_h_row_irfft512<<<256 * L,   T, 0, stream>>>(T1, ht);
    k_h_row_fft1024 <<<256 * L,   T, 0, stream>>>(ht, T2);
    k_h_col_fft1024 <<<256 * NF,  T, 0, stream>>>(T2, h0, Hre, Him);

    // X path
    k_x_row_fft<<<Bb * L  * D, T, 0, stream>>>(x, Xrow);
    k_x_col_fft<<<Bb * NF * D, T, 0, stream>>>(Xrow, Xsre, Xsim);

    // per-frequency channel mix (WMMA fp32 + TDM); 8 frequencies per block
    k_mix_wmma<<<(N2 * NF) / 8, T, 0, stream>>>(Xsre, Xsim, Hre, Him, Ysre, Ysim);

    // Y inverse path
    k_y_col_ifft<<<Bb * NF * D, T, 0, stream>>>(Ysre, Ysim, Ycol);
    k_y_row_ifft<<<Bb * L  * D, T, 0, stream>>>(Ycol, out);
}